// AdditiveAttention_25125558682350
// MI455X (gfx1250) — compile-verified
//
#include <hip/hip_runtime.h>
#include <hip/hip_bf16.h>

// Problem constants (from reference setup_inputs)
#define BB 4      // batch
#define QQ 256    // queries
#define KK 1024   // keys
#define EE 256    // embedding (q/k)
#define HH 128    // hidden h
#define VV 256    // value embedding

typedef __attribute__((ext_vector_type(2))) float v2f;
typedef __attribute__((ext_vector_type(8))) float v8f;

__device__ __forceinline__ float dev_tanh(float x) {
#if __has_builtin(__builtin_amdgcn_tanhf)
    return __builtin_amdgcn_tanhf(x);          // V_TANH_F32 (gfx1250 TRANS unit)
#else
    float xx = fminf(fmaxf(x, -15.f), 15.f);
    float t  = __expf(2.0f * xx);
    return (t - 1.0f) / (t + 1.0f);
#endif
}

// ---------------------------------------------------------------------------
// Generic batched GEMM  C[M,N] = A[M,K] * B[K,N]   (all row-major, fp32)
// via V_WMMA_F32_16X16X4_F32.  One 16x16 output tile per wave, 4 waves/block.
// A 16x4 layout : lane L holds row M=L%16, K pair {koff, koff+1}, koff=2*(L/16)
// B 4x16 layout : lane L holds col N=L%16, K pair {koff, koff+1}
// C/D 16x16     : vgpr r -> M = 8*(L/16)+r, N = L%16
// ---------------------------------------------------------------------------
__global__ __launch_bounds__(128)
void wmma_gemm_f32(const float* __restrict__ A, const float* __restrict__ Bm,
                   float* __restrict__ C,
                   int M, int N, int Kd, int lda, int ldb, int ldc,
                   long long Abs, long long Bbs, long long Cbs) {
    const int batch = blockIdx.z;
    A  += (size_t)batch * Abs;
    Bm += (size_t)batch * Bbs;
    C  += (size_t)batch * Cbs;

    const int wave = threadIdx.x >> 5;
    const int lane = threadIdx.x & 31;
    const int tilesN = N >> 4;

    const int tile = blockIdx.x * (blockDim.x >> 5) + wave;
    const int tm = (tile / tilesN) << 4;
    const int tn = (tile % tilesN) << 4;
    if (tm >= M) return;                      // uniform per wave: EXEC stays full

    const int l16  = lane & 15;
    const int half = lane >> 4;               // 0 | 1
    const int koff = half << 1;               // 0 | 2

    const float* arow = A  + (size_t)(tm + l16) * lda;
    const float* bcol = Bm + (tn + l16);

    v8f c = {};
#pragma unroll 4
    for (int kk = 0; kk < Kd; kk += 4) {
        v2f av;
        av.x = arow[kk + koff];
        av.y = arow[kk + koff + 1];
        v2f bv;
        bv.x = bcol[(size_t)(kk + koff) * ldb];
        bv.y = bcol[(size_t)(kk + koff + 1) * ldb];
        c = __builtin_amdgcn_wmma_f32_16x16x4_f32(
                false, av, false, bv, (short)0, c, false, false);
    }

    float* crow = C + (size_t)(tm + (half << 3)) * ldc + tn + l16;
#pragma unroll
    for (int r = 0; r < 8; ++r)
        crow[(size_t)r * ldc] = c[r];
}

// ---------------------------------------------------------------------------
// Fused scores + softmax:  one block (256 thr, 8 waves) per (b,q) row.
//   z[k]  = sum_h wv[h] * tanh(Hq[b,q,h] + Hk[b,k,h])
//   a[k]  = softmax_k(z)
// Each thread owns 4 k values; reductions via wave shuffles + LDS.
// ---------------------------------------------------------------------------
__global__ __launch_bounds__(256)
void scores_softmax(const float* __restrict__ Hq, const float* __restrict__ Hk,
                    const float* __restrict__ wv, float* __restrict__ a_out) {
    const int row = blockIdx.x;               // b*QQ + q
    const int b   = row >> 8;                 // QQ == 256
    const int tid = threadIdx.x;
    const int lane = tid & 31;
    const int wid  = tid >> 5;

    __shared__ float s_hq[HH];
    __shared__ float s_wv[HH];
    __shared__ float s_max[8];
    __shared__ float s_sum[8];

    if (tid < HH) {
        s_hq[tid] = Hq[(size_t)row * HH + tid];
        s_wv[tid] = wv[tid];
    }
    __syncthreads();

    float z[4];
#pragma unroll
    for (int i = 0; i < 4; ++i) {
        const int k = tid + (i << 8);
        const float4* hk = (const float4*)(Hk + ((size_t)b * KK + k) * HH);
        float acc = 0.f;
#pragma unroll 8
        for (int h4 = 0; h4 < HH / 4; ++h4) {
            float4 hv = hk[h4];
            const int h = h4 << 2;
            acc = fmaf(s_wv[h + 0], dev_tanh(s_hq[h + 0] + hv.x), acc);
            acc = fmaf(s_wv[h + 1], dev_tanh(s_hq[h + 1] + hv.y), acc);
            acc = fmaf(s_wv[h + 2], dev_tanh(s_hq[h + 2] + hv.z), acc);
            acc = fmaf(s_wv[h + 3], dev_tanh(s_hq[h + 3] + hv.w), acc);
        }
        z[i] = acc;
    }

    // ---- row max ----
    float lm = fmaxf(fmaxf(z[0], z[1]), fmaxf(z[2], z[3]));
#pragma unroll
    for (int off = 16; off > 0; off >>= 1)
        lm = fmaxf(lm, __shfl_xor(lm, off, 32));
    if (lane == 0) s_max[wid] = lm;
    __syncthreads();
    float rowmax = s_max[0];
#pragma unroll
    for (int j = 1; j < 8; ++j) rowmax = fmaxf(rowmax, s_max[j]);

    // ---- exp + sum ----
    float e[4];
    float ls = 0.f;
#pragma unroll
    for (int i = 0; i < 4; ++i) {
        e[i] = __expf(z[i] - rowmax);
        ls += e[i];
    }
#pragma unroll
    for (int off = 16; off > 0; off >>= 1)
        ls += __shfl_xor(ls, off, 32);
    if (lane == 0) s_sum[wid] = ls;
    __syncthreads();
    float tot = 0.f;
#pragma unroll
    for (int j = 0; j < 8; ++j) tot += s_sum[j];
    const float inv = 1.0f / tot;

    float* arow = a_out + (size_t)row * KK;
#pragma unroll
    for (int i = 0; i < 4; ++i)
        arow[tid + (i << 8)] = e[i] * inv;
}

extern "C" void kernel_launch(void* const* d_in, const int* in_sizes, int n_in,
                              void* d_out, int out_size, void* d_ws, size_t ws_size,
                              hipStream_t stream) {
    const float* query = (const float*)d_in[0];   // (4,256,256)
    const float* key   = (const float*)d_in[1];   // (4,1024,256)
    const float* value = (const float*)d_in[2];   // (4,1024,256)
    const float* wq    = (const float*)d_in[3];   // (256,128)
    const float* wk    = (const float*)d_in[4];   // (256,128)
    const float* wv    = (const float*)d_in[5];   // (128,)

    float* out = (float*)d_out;                   // (4,256,256)
    float* a   = out + (size_t)BB * QQ * VV;      // (4,256,1024)

    float* Hq = (float*)d_ws;                     // (4*256, 128)
    float* Hk = Hq + (size_t)BB * QQ * HH;        // (4*1024, 128)

    // Hq = query(1024x256) @ wq(256x128): 512 tiles -> 128 blocks x 4 waves
    wmma_gemm_f32<<<dim3(128, 1, 1), 128, 0, stream>>>(
        query, wq, Hq, BB * QQ, HH, EE, EE, HH, HH, 0, 0, 0);

    // Hk = key(4096x256) @ wk(256x128): 2048 tiles -> 512 blocks
    wmma_gemm_f32<<<dim3(512, 1, 1), 128, 0, stream>>>(
        key, wk, Hk, BB * KK, HH, EE, EE, HH, HH, 0, 0, 0);

    // fused tanh-scores + softmax: one block per (b,q) row
    scores_softmax<<<dim3(BB * QQ, 1, 1), 256, 0, stream>>>(Hq, Hk, wv, a);

    // out[b] = a[b](256x1024) @ value[b](1024x256): 256 tiles/batch -> 64 blocks, z=4
    wmma_gemm_f32<<<dim3(64, 1, BB), 128, 0, stream>>>(
        a, value, out, QQ, VV, KK, KK, VV, VV,
        (long long)QQ * KK, (long long)KK * VV, (long long)QQ * VV);
}